// FGL_48155173323481
// MI455X (gfx1250) — compile-verified
//
#include <hip/hip_runtime.h>
#include <hip/hip_bf16.h>

// Problem constants (match reference)
#define NB    16      // N
#define INC   32
#define INN   65536
#define OUTC  32
#define OUTN  8192
#define MAXD  16

typedef float v2f __attribute__((ext_vector_type(2)));
typedef float v8f __attribute__((ext_vector_type(8)));

// ---------------------------------------------------------------------------
// Kernel 1: g[i][n][d] = sum_c x[n][c][i] * weight[c][i] * ct_weight[d][c]
// Per block: fixed n, 64 consecutive i. 128 threads = 4 waves.
// Each wave: M=16 (i rows), N=32 (two 16-wide d tiles), K=32 (8 x K=4 WMMA).
// ---------------------------------------------------------------------------
__global__ __launch_bounds__(128) void FGL_build_g(
    const float* __restrict__ x,       // (NB, INC, INN)
    const float* __restrict__ weight,  // (INC, INN)
    const float* __restrict__ ctw,     // (OUTC, INC)
    float* __restrict__ g)             // (INN, NB, OUTC)
{
  __shared__ float xl[64 * 33];        // [i][c], padded stride 33

  const int n   = blockIdx.y;
  const int i0  = blockIdx.x * 64;
  const int tid = threadIdx.x;

  // Coalesced load of x tile, scaled by weight: xl[i][c] = x[n][c][i0+i]*w[c][i0+i]
  const float* xb = x + ((size_t)n * INC) * INN + i0;
  const float* wb = weight + i0;
#pragma unroll
  for (int r = 0; r < 16; ++r) {
    int idx = r * 128 + tid;         // 2048 elements total
    int c = idx >> 6;                // 0..31
    int i = idx & 63;                // 0..63
    xl[i * 33 + c] = xb[(size_t)c * INN + i] * wb[(size_t)c * INN + i];
  }
  __syncthreads();

  const int wave  = tid >> 5;        // 0..3 -> i rows 16*wave..
  const int lane  = tid & 31;
  const int lhalf = lane & 15;
  const int hi    = lane >> 4;       // 0: K0/K1 pair, 1: K2/K3 pair
  const int m     = wave * 16 + lhalf;

  v8f acc0 = {};                     // d tile 0..15
  v8f acc1 = {};                     // d tile 16..31
#pragma unroll
  for (int k = 0; k < 8; ++k) {
    const int c0 = k * 4 + hi * 2;   // absolute K for v0 of this lane half
    // A fragment (16x4 f32): lanes 0-15: M=lane, v0=K0,v1=K1; lanes 16-31: K2,K3
    v2f a;
    a.x = xl[m * 33 + c0];
    a.y = xl[m * 33 + c0 + 1];
    // B fragment (4x16 f32): B[kk][d] = ct_weight[d][kk]; lane half selects K pair
    v2f b0, b1;
    b0.x = ctw[lhalf * INC + c0];
    b0.y = ctw[lhalf * INC + c0 + 1];
    b1.x = ctw[(lhalf + 16) * INC + c0];
    b1.y = ctw[(lhalf + 16) * INC + c0 + 1];
    acc0 = __builtin_amdgcn_wmma_f32_16x16x4_f32(false, a, false, b0,
                                                 (short)0, acc0, false, false);
    acc1 = __builtin_amdgcn_wmma_f32_16x16x4_f32(false, a, false, b1,
                                                 (short)0, acc1, false, false);
  }

  // D layout (16x16 f32): VGPR r -> lanes 0-15: M=r, N=lhalf; lanes 16-31: M=r+8
  float* gb = g + ((size_t)(i0 + wave * 16) * NB + n) * OUTC;
#pragma unroll
  for (int r = 0; r < 8; ++r) {
    const int M = r + hi * 8;
    gb[(size_t)M * (NB * OUTC) + lhalf]      = acc0[r];
    gb[(size_t)M * (NB * OUTC) + lhalf + 16] = acc1[r];
  }
}

// ---------------------------------------------------------------------------
// Kernel 2: out[n][d][o] = sum_j wj[o][j]*g[A[o][j]][n][d] + ct_bias[d] + bias[d][o]
// 512 threads = full (n,d) plane; OTILE output nodes per block.
// ---------------------------------------------------------------------------
#define OTILE 16
#define APAD  516   // LDS row stride (516 % 64 == 4 -> conflict-free transpose)

__global__ __launch_bounds__(512) void FGL_gather_pool(
    const float* __restrict__ g,        // (INN, NB, OUTC)
    const int*   __restrict__ A,        // (OUTN, MAXD)
    const float* __restrict__ mask,     // (OUTN, MAXD)
    const float* __restrict__ mw,       // (OUTN, MAXD)
    const float* __restrict__ ctb,      // (OUTC,)
    const float* __restrict__ bias,     // (OUTC, OUTN)
    float* __restrict__ out)            // (NB, OUTC, OUTN)
{
  __shared__ float acc_lds[OTILE * APAD];
  __shared__ int   aidx[OTILE * MAXD];
  __shared__ float wj[OTILE * MAXD];

  const int tid = threadIdx.x;         // 0..511 == n*32 + d
  const int o0  = blockIdx.x * OTILE;

  // Stage neighbor indices and combined weights (256 entries)
  if (tid < OTILE * MAXD) {
    const size_t e = (size_t)o0 * MAXD + tid;
    aidx[tid] = A[e];
    wj[tid]   = mw[e] * mask[e];
  }
  __syncthreads();

  const int d = tid & 31;
  const float cb = ctb[d];

  for (int o = 0; o < OTILE; ++o) {
    float acc = cb;
#pragma unroll
    for (int j = 0; j < MAXD; ++j) {
      const int   a = aidx[o * MAXD + j];
      const float w = wj[o * MAXD + j];
      acc += w * g[(size_t)a * (NB * OUTC) + tid];   // coalesced 2KB row
    }
    acc_lds[o * APAD + tid] = acc;
  }
  __syncthreads();

  // Transposed writeout: consecutive threads -> consecutive o (coalesced),
  // fuse bias[d][o].
  for (int pass = 0; pass < OTILE; ++pass) {
    const int v  = pass * 512 + tid;
    const int o  = v & (OTILE - 1);
    const int nd = v >> 4;             // log2(OTILE)
    const int dd = nd & 31;
    out[(size_t)nd * OUTN + o0 + o] =
        acc_lds[o * APAD + nd] + bias[(size_t)dd * OUTN + o0 + o];
  }
}

// ---------------------------------------------------------------------------
extern "C" void kernel_launch(void* const* d_in, const int* in_sizes, int n_in,
                              void* d_out, int out_size, void* d_ws, size_t ws_size,
                              hipStream_t stream) {
  const float* x      = (const float*)d_in[0];
  const int*   A      = (const int*)  d_in[1];
  const float* weight = (const float*)d_in[2];
  const float* mask   = (const float*)d_in[3];
  const float* mw     = (const float*)d_in[4];
  const float* ctw    = (const float*)d_in[5];
  const float* ctb    = (const float*)d_in[6];
  const float* bias   = (const float*)d_in[7];
  float*       out    = (float*)d_out;
  float*       g      = (float*)d_ws;   // 65536*16*32 f32 = 128 MiB scratch

  dim3 grid1(INN / 64, NB);
  FGL_build_g<<<grid1, 128, 0, stream>>>(x, weight, ctw, g);

  FGL_gather_pool<<<OUTN / OTILE, 512, 0, stream>>>(g, A, mask, mw, ctb, bias, out);
}